// EC_MoEGate_29729763623448
// MI455X (gfx1250) — compile-verified
//
#include <hip/hip_runtime.h>
#include <hip/hip_bf16.h>

// Problem constants (from reference): B=4, S=4096, d=2048, E=16, capacity=512
#define B_DIM 4
#define S_DIM 4096
#define D_DIM 2048
#define E_DIM 16
#define CAP   512
#define T_TOT (B_DIM * S_DIM)   // 16384 tokens

typedef float v2f __attribute__((ext_vector_type(2)));
typedef float v8f __attribute__((ext_vector_type(8)));

// ---------------------------------------------------------------------------
// Kernel 1: gate logits via V_WMMA_F32_16X16X4_F32 + softmax over E=16.
//   A (M x K) = weight  : M=16 experts, K=d   (fp32, 2 VGPRs / lane / step)
//   B (K x N) = hidden  : N=16 tokens
//   D (M x N) : lane n(0..15) = token n; VGPR i = expert i (lanes 0-15)
//               or expert i+8 (lanes 16-31).
//
// Data movement (MI455X, 23.3 TB/s HBM; kernel is a pure HBM stream of X):
//  * hidden chunk (16 tokens x 32 K): direct global_load_b128 -> ds_store_b128
//    staging (the scheduler clauses the 4 loads and retires stores with
//    incremental s_wait_loadcnt; no register double-buffer -> no scratch
//    spills).
//  * weight (128 KB) staged ONCE per WG into LDS, padded row stride 2052
//    dwords (banks 4e+k distinct across the 16 expert lanes; the +2 K-offset
//    of lanes 16-31 cannot collide mod 4) -> conflict-free ds_load_b64 frags.
//  * one branchless global_prefetch_b8 per chunk: 32 lanes cover the 32
//    token rows of the chunks two and three steps ahead.
//  Dynamic LDS: 16*2052*4 + 8*16*36*4 = 149,760 B -> 2 WGs per 320 KB WGP.
// ---------------------------------------------------------------------------
#define CHUNK_K   32
#define XS_STRIDE 36           // 32 + 4 pad dwords
#define WS_STRIDE 2052         // 2048 + 4 pad dwords
#define WAVES_PER_WG 8
#define LDS_BYTES ((E_DIM * WS_STRIDE + WAVES_PER_WG * 16 * XS_STRIDE) * 4)

__global__ __launch_bounds__(256, 1)
void moe_gate_scores_kernel(const float* __restrict__ X,
                            const float* __restrict__ W,
                            float* __restrict__ scores)
{
    extern __shared__ float lds[];
    float* Ws = lds;                                        // 16 x WS_STRIDE
    const int lane   = threadIdx.x & 31;
    const int waveId = threadIdx.x >> 5;
    float* xs = lds + E_DIM * WS_STRIDE + waveId * (16 * XS_STRIDE);

    // ---- stage full weight matrix into LDS (coalesced, once per WG)
    for (int i = threadIdx.x; i < E_DIM * (D_DIM / 4); i += 256) {
        const int e = i >> 9;                // / 512
        const int q = i & 511;
        const float4 v = *(const float4*)(W + (size_t)e * D_DIM + 4 * q);
        *(float4*)&Ws[e * WS_STRIDE + 4 * q] = v;
    }
    __syncthreads();

    const int tile_base = blockIdx.x * (WAVES_PER_WG * 16) + waveId * 16;
    const int lm  = lane & 15;               // token (B-frag) / expert (A-frag)
    const int khi = (lane >> 4) << 1;        // +0 lanes 0-15, +2 lanes 16-31

    v8f acc = {};

    for (int k0 = 0; k0 < D_DIM; k0 += CHUNK_K) {
        // ---- stage 16 tokens x CHUNK_K floats into this wave's LDS region.
        // 128 float4 over 32 lanes x 4 iters; fully coalesced per token row.
        #pragma unroll
        for (int it = 0; it < 4; ++it) {
            const int f = lane + 32 * it;    // [0,128)
            const int t = f >> 3;            // token within tile
            const int q = f & 7;             // float4 column
            const float4 v = *(const float4*)(X + (size_t)(tile_base + t) * D_DIM
                                                + k0 + 4 * q);
            *(float4*)&xs[t * XS_STRIDE + 4 * q] = v;
        }

        // ---- one branchless prefetch per chunk: 32 lanes = 32 token rows
        // of the chunks two and three steps ahead (wrapped on the tail, no branch).
        int kp = k0 + 2 * CHUNK_K;
        kp = (kp < D_DIM) ? kp : 0;          // uniform select
        __builtin_prefetch(X + (size_t)(tile_base + (lane >> 1)) * D_DIM
                             + kp + (lane & 1) * 32, 0, 3);

        // ---- 8 WMMA steps of K=4, all operands from LDS
        // (same-wave DS is in-order: fragment loads see this chunk's stores,
        //  and next iteration's stores are ordered after these loads)
        #pragma unroll
        for (int kk = 0; kk < CHUNK_K; kk += 4) {
            const int koff = kk + khi;       // VGPR0 holds K=koff, VGPR1 K=koff+1
            v2f a = *(const v2f*)&Ws[lm * WS_STRIDE + k0 + koff]; // weight frag
            v2f b = *(const v2f*)&xs[lm * XS_STRIDE + koff];      // hidden frag
            acc = __builtin_amdgcn_wmma_f32_16x16x4_f32(
                      /*neg_a=*/false, a, /*neg_b=*/false, b,
                      /*c_mod=*/(short)0, acc,
                      /*reuse_a=*/false, /*reuse_b=*/false);
        }
    }

    // ---- softmax over E=16: 8 logits local + 8 logits in lane^16
    float m = acc[0];
    #pragma unroll
    for (int i = 1; i < 8; ++i) m = fmaxf(m, acc[i]);
    float mo = __shfl_xor(m, 16, 32);
    const float M = fmaxf(m, mo);

    float e[8];
    float s = 0.0f;
    #pragma unroll
    for (int i = 0; i < 8; ++i) { e[i] = __expf(acc[i] - M); s += e[i]; }
    const float so  = __shfl_xor(s, 16, 32);
    const float inv = 1.0f / (s + so);

    // ---- store expert-major scores: scores[(b*E + e) * S + s]
    const int tok = tile_base + lm;
    const int bb  = tok >> 12;               // / S_DIM
    const int ss  = tok & (S_DIM - 1);
    const int e0  = (lane < 16) ? 0 : 8;
    #pragma unroll
    for (int i = 0; i < 8; ++i) {
        scores[((size_t)(bb * E_DIM + e0 + i) << 12) + ss] = e[i] * inv;
    }
}

// ---------------------------------------------------------------------------
// Kernel 2: per-(b,e) full bitonic sort of 4096 scores in LDS (descending,
// index-ascending tiebreak, matching jax.lax.top_k ordering), emit top-512.
// One block per row; 256 threads; 32 KB LDS; source is L2-resident scratch.
// ---------------------------------------------------------------------------
__device__ __forceinline__ bool before_desc(float av, int ai, float bv, int bi) {
    return (av > bv) || (av == bv && ai < bi);
}

__global__ __launch_bounds__(256)
void moe_topk_kernel(const float* __restrict__ scores,
                     long long* __restrict__ out_idx,
                     float* __restrict__ out_w)
{
    __shared__ float sv[S_DIM];
    __shared__ int   si[S_DIM];

    const int row = blockIdx.x;              // (b*E + e)
    const int tid = threadIdx.x;
    const float* src = scores + (size_t)row * S_DIM;

    for (int i = tid; i < S_DIM; i += 256) { sv[i] = src[i]; si[i] = i; }
    __syncthreads();

    for (int k = 2; k <= S_DIM; k <<= 1) {
        for (int j = k >> 1; j > 0; j >>= 1) {
            for (int t = tid; t < (S_DIM / 2); t += 256) {
                const int i   = ((t & ~(j - 1)) << 1) | (t & (j - 1));
                const int ixj = i | j;
                const bool up = ((i & k) == 0);          // descending region
                float av = sv[i],   bv = sv[ixj];
                int   ai = si[i],   bi = si[ixj];
                const bool aFirst = before_desc(av, ai, bv, bi);
                if (up ? !aFirst : aFirst) {
                    sv[i] = bv; sv[ixj] = av;
                    si[i] = bi; si[ixj] = ai;
                }
            }
            __syncthreads();
        }
    }

    for (int i = tid; i < CAP; i += 256) {
        out_idx[(size_t)row * CAP + i] = (long long)si[i];
        out_w  [(size_t)row * CAP + i] = sv[i];
    }
}

// ---------------------------------------------------------------------------
extern "C" void kernel_launch(void* const* d_in, const int* in_sizes, int n_in,
                              void* d_out, int out_size, void* d_ws, size_t ws_size,
                              hipStream_t stream)
{
    const float* X = (const float*)d_in[0];  // (B, S, d) fp32
    const float* W = (const float*)d_in[1];  // (E, d)   fp32

    // Scratch: expert-major scores (B, E, S) fp32 = 1 MiB (fits in L2).
    float* scores = (float*)d_ws;

    // Output layout: tuple(int64 idx (B,E,CAP), f32 weight (B,E,CAP)),
    // concatenated flat in return order.
    long long* out_idx = (long long*)d_out;
    float*     out_w   = (float*)((char*)d_out +
                                  (size_t)B_DIM * E_DIM * CAP * sizeof(long long));

    // Kernel 1: 16384 tokens / (8 waves * 16 tokens) = 128 workgroups,
    // 149,760 B dynamic LDS each (2 WGs per 320 KB WGP).
    moe_gate_scores_kernel<<<dim3(T_TOT / (WAVES_PER_WG * 16)), dim3(256),
                             LDS_BYTES, stream>>>(X, W, scores);

    // Kernel 2: one block per (b, e) row = 64 blocks.
    moe_topk_kernel<<<dim3(B_DIM * E_DIM), dim3(256), 0, stream>>>(
        scores, out_idx, out_w);
}